// SNNLayer_36077725286942
// MI455X (gfx1250) — compile-verified
//
#include <hip/hip_runtime.h>

#define NN 1024
#define OO 1024
#define BB 64
#define BIGV 1e10f

typedef float v2f __attribute__((ext_vector_type(2)));
typedef float v8f __attribute__((ext_vector_type(8)));

// ---------------------------------------------------------------------------
// Kernel 1: transpose w [O,N] -> wT [N,O] so column gathers become contiguous.
// ---------------------------------------------------------------------------
__global__ __launch_bounds__(256) void transpose_w(const float* __restrict__ w,
                                                   float* __restrict__ wT) {
    __shared__ float tile[32][33];
    const int bn = blockIdx.x * 32;   // N tile base
    const int bo = blockIdx.y * 32;   // O tile base
    const int tx = threadIdx.x;       // 0..31
    const int ty = threadIdx.y;       // 0..7
    #pragma unroll
    for (int i = ty; i < 32; i += 8)
        tile[i][tx] = w[(size_t)(bo + i) * NN + bn + tx];
    __syncthreads();
    #pragma unroll
    for (int i = ty; i < 32; i += 8)
        wT[(size_t)(bn + i) * OO + bo + tx] = tile[tx][i];
}

// ---------------------------------------------------------------------------
// Kernel 2: per-batch-row bitonic argsort of x (1024 keys, 512 threads).
// ---------------------------------------------------------------------------
__global__ __launch_bounds__(512) void sort_x(const float* __restrict__ x,
                                              float* __restrict__ xs,
                                              int* __restrict__ order) {
    __shared__ float sk[NN];
    __shared__ int   sv[NN];
    const int b = blockIdx.x;
    const int t = threadIdx.x;
    sk[t]       = x[(size_t)b * NN + t];        sv[t]       = t;
    sk[t + 512] = x[(size_t)b * NN + t + 512];  sv[t + 512] = t + 512;

    for (int size = 2; size < NN; size <<= 1) {
        const int ddd = ((t & (size >> 1)) != 0) ? 0 : 1;
        for (int stride = size >> 1; stride > 0; stride >>= 1) {
            __syncthreads();
            const int pos = 2 * t - (t & (stride - 1));
            float ka = sk[pos], kb = sk[pos + stride];
            int   va = sv[pos], vb = sv[pos + stride];
            if (((ka > kb) ? 1 : 0) == ddd) {
                sk[pos] = kb; sk[pos + stride] = ka;
                sv[pos] = vb; sv[pos + stride] = va;
            }
        }
    }
    for (int stride = NN / 2; stride > 0; stride >>= 1) {
        __syncthreads();
        const int pos = 2 * t - (t & (stride - 1));
        float ka = sk[pos], kb = sk[pos + stride];
        int   va = sv[pos], vb = sv[pos + stride];
        if (ka > kb) {
            sk[pos] = kb; sk[pos + stride] = ka;
            sv[pos] = vb; sv[pos + stride] = va;
        }
    }
    __syncthreads();
    xs[(size_t)b * NN + t]          = sk[t];
    xs[(size_t)b * NN + t + 512]    = sk[t + 512];
    order[(size_t)b * NN + t]       = sv[t];
    order[(size_t)b * NN + t + 512] = sv[t + 512];
}

// ---------------------------------------------------------------------------
// Kernel 3: WMMA prefix scan, double-buffered async staging, early exit.
// Grid: (O/64, B). Block: 128 threads = 4 waves; each wave owns 16 outputs.
// ---------------------------------------------------------------------------
#define STRIDE 68   // floats per staged row: 272 B, 16B-aligned, bank-skewed

__global__ __launch_bounds__(128) void snn_scan(const float* __restrict__ wT,
                                                const float* __restrict__ xs,
                                                const int* __restrict__ order,
                                                float* __restrict__ out) {
    __shared__ float stage[2][16 * STRIDE];
    __shared__ float xsStage[2][16];
    __shared__ int   doneCnt;

    const int b     = blockIdx.y;
    const int o_blk = blockIdx.x * 64;
    const int t     = threadIdx.x;      // 0..127
    const int lane  = t & 31;
    const int wave  = t >> 5;           // 0..3
    const int n     = lane & 15;        // column (k-position) / A row index
    const int khalf = lane >> 4;        // 0: lanes 0-15, 1: lanes 16-31
    const int wo    = wave * 16;        // wave's o offset in the 64-window

    if (t == 0) doneCnt = 0;

    // staging mapping: thread t copies 32 B of row s_kk (two b128 asyncs)
    const int s_kk = t >> 3;            // 0..15
    const int s_o8 = (t & 7) * 8;       // 0,8,...,56
    const unsigned lds_dst0 =
        (unsigned)(unsigned long long)(void*)&stage[0][s_kk * STRIDE + s_o8];
    const unsigned lds_dst1 =
        (unsigned)(unsigned long long)(void*)&stage[1][s_kk * STRIDE + s_o8];
    const unsigned lds_xs0 =
        (unsigned)(unsigned long long)(void*)&xsStage[0][t & 15];
    const unsigned lds_xs1 =
        (unsigned)(unsigned long long)(void*)&xsStage[1][t & 15];

    float carryW[8], carryWX[8];
    #pragma unroll
    for (int r = 0; r < 8; ++r) { carryW[r] = 0.f; carryWX[r] = 0.f; }
    unsigned doneMask = 0;              // uniform within each 16-lane half

    auto stageChunk = [&](int c, int buf) {
        const int j = order[(size_t)b * NN + c * 16 + s_kk];
        unsigned long long g =
            (unsigned long long)(const void*)&wT[(size_t)j * OO + o_blk + s_o8];
        const unsigned ld = buf ? lds_dst1 : lds_dst0;
        asm volatile("global_load_async_to_lds_b128 %0, %1, off"
                     :: "v"(ld), "v"(g) : "memory");
        asm volatile("global_load_async_to_lds_b128 %0, %1, off offset:16"
                     :: "v"(ld), "v"(g) : "memory");
        if (t < 16) {
            unsigned long long gx =
                (unsigned long long)(const void*)&xs[(size_t)b * NN + c * 16 + t];
            asm volatile("global_load_async_to_lds_b32 %0, %1, off"
                         :: "v"(buf ? lds_xs1 : lds_xs0), "v"(gx) : "memory");
        }
    };

    __syncthreads();                    // doneCnt initialized
    stageChunk(0, 0);                   // prologue: fill buffer 0

    for (int c = 0; c < NN / 16; ++c) {
        const int buf = c & 1;
        asm volatile("s_wait_asynccnt 0" ::: "memory");
        __syncthreads();                // buffer `buf` ready for everyone
        if (doneCnt == 4) break;        // uniform: read after barrier
        if (c + 1 < NN / 16) stageChunk(c + 1, buf ^ 1);   // prefetch behind compute

        const bool halfDoneBefore = (doneMask == 0xFFu);
        if (!__all(halfDoneBefore)) {   // wave-uniform: EXEC stays full for WMMA
            const float xsv = xsStage[buf][n];

            v8f cw  = {};
            v8f cwx = {};
            #pragma unroll
            for (int kg = 0; kg < 4; ++kg) {
                const int j0 = kg * 4 + khalf * 2;      // A/B VGPR K mapping
                const float a0 = stage[buf][j0 * STRIDE + wo + n];
                const float a1 = stage[buf][(j0 + 1) * STRIDE + wo + n];
                const float x0 = xsStage[buf][j0];
                const float x1 = xsStage[buf][j0 + 1];
                v2f A;  A.x  = a0;       A.y  = a1;
                v2f AX; AX.x = a0 * x0;  AX.y = a1 * x1;
                v2f B;  B.x  = (j0     <= n) ? 1.f : 0.f;   // L lower-triangular
                        B.y  = (j0 + 1 <= n) ? 1.f : 0.f;
                cw  = __builtin_amdgcn_wmma_f32_16x16x4_f32(
                          false, A,  false, B, (short)0, cw,  false, false);
                cwx = __builtin_amdgcn_wmma_f32_16x16x4_f32(
                          false, AX, false, B, (short)0, cwx, false, false);
            }

            unsigned newly = 0;
            #pragma unroll
            for (int r = 0; r < 8; ++r) {
                const float cumW  = carryW[r]  + cw[r];
                const float cumWX = carryWX[r] + cwx[r];
                const float denom = fminf(fmaxf(cumW - 1.0f, 1e-10f), 1e10f);
                // denom > 0, so (cumWX/denom > xs) <=> (cumWX > xs*denom):
                const bool cond = (cumW > 1.0f) && (cumWX > xsv * denom) &&
                                  !((doneMask >> r) & 1u);
                const unsigned m32  = (unsigned)__ballot(cond);
                const unsigned half = khalf ? (m32 >> 16) : (m32 & 0xFFFFu);
                const int cand = half ? (int)__builtin_ctz(half) : 32;
                if (cand < 16) {
                    newly |= (1u << r);
                    // the winning lane already holds cumWX/denom in registers:
                    if (n == cand)   // exact division, once per output row
                        out[(size_t)b * OO + o_blk + wo + khalf * 8 + r] =
                            cumWX / denom;
                }
                carryW[r]  += __shfl(cw[r],  15 + (khalf << 4));   // chunk totals
                carryWX[r] += __shfl(cwx[r], 15 + (khalf << 4));
            }
            doneMask |= newly;
            if (__all(doneMask == 0xFFu) && lane == 0) atomicAdd(&doneCnt, 1);
        }
    }

    // rows that never fired take the sentinel (idx == N path of the reference)
    if (n == 0) {
        #pragma unroll
        for (int r = 0; r < 8; ++r)
            if (!((doneMask >> r) & 1u))
                out[(size_t)b * OO + o_blk + wo + khalf * 8 + r] = BIGV;
    }
}

// ---------------------------------------------------------------------------
extern "C" void kernel_launch(void* const* d_in, const int* in_sizes, int n_in,
                              void* d_out, int out_size, void* d_ws, size_t ws_size,
                              hipStream_t stream) {
    const float* x = (const float*)d_in[0];   // [B, N]
    const float* w = (const float*)d_in[1];   // [O, N]
    float* out = (float*)d_out;               // [B, O, 1] flat

    char* ws = (char*)d_ws;
    float* wT    = (float*)ws;                                        // 4 MB
    float* xsrt  = (float*)(ws + (size_t)NN * OO * 4);                // 256 KB
    int*   order = (int*)  (ws + (size_t)NN * OO * 4 + (size_t)BB * NN * 4);

    transpose_w<<<dim3(NN / 32, OO / 32), dim3(32, 8), 0, stream>>>(w, wT);
    sort_x<<<BB, 512, 0, stream>>>(x, xsrt, order);
    snn_scan<<<dim3(OO / 64, BB), 128, 0, stream>>>(wT, xsrt, order, out);
}